// BoxPairMultiScaleRoIAlign_19662360281584
// MI455X (gfx1250) — compile-verified
//
#include <hip/hip_runtime.h>
#include <hip/hip_bf16.h>

typedef float v2f __attribute__((ext_vector_type(2)));
typedef float v8f __attribute__((ext_vector_type(8)));

#define SS 14           // P*SR sample points per axis
#define NTAP_CHUNK 28   // 7 samples * 4 taps per chunk
#define TROW 264        // padded LDS row stride (floats); 2*264 % 64 == 16 -> conflict-free halves
#define CCH 256

__device__ __forceinline__ float coverage_at(float x1, float x2, float cf) {
    // fractional per-pixel coverage of [x1,x2] at integer column cf
    float x1f = floorf(x1);
    float x2c = ceilf(x2);
    float inside = (cf >= x1f && cf < x2c) ? 1.0f : 0.0f;
    float wl = (cf == x1f) ? (1.0f + x1f - x1) : 1.0f;
    float wr = (cf == x2c - 1.0f) ? (1.0f + x2 - x2c) : 1.0f;
    return inside * wl * wr;
}

__device__ __forceinline__ void axis_interp(float coord, float fs, int& lo, int& hi,
                                            float& wl, float& wh) {
    // torchvision roi_align (aligned=False) 1D bilinear setup
    float valid = (coord >= -1.0f && coord <= fs) ? 1.0f : 0.0f;
    float c = fmaxf(coord, 0.0f);
    float lo0 = floorf(c);
    bool top = (lo0 >= fs - 1.0f);
    float lof = top ? (fs - 1.0f) : lo0;
    float hif = top ? (fs - 1.0f) : (lo0 + 1.0f);
    c = top ? (fs - 1.0f) : c;
    float frac = c - lof;
    lo = (int)lof;
    hi = (int)hif;
    wl = (1.0f - frac) * valid;
    wh = frac * valid;
}

__global__ __launch_bounds__(256)
void boxpair_roialign_wmma(const float* __restrict__ f0, const float* __restrict__ f1,
                           const float* __restrict__ f2, const float* __restrict__ f3,
                           const float* __restrict__ bxh, const float* __restrict__ bxo,
                           float* __restrict__ out, int K) {
    __shared__ int   s_xlo[SS], s_xhi[SS], s_ylo[SS], s_yhi[SS];
    __shared__ float s_wx[2][SS], s_wy[2][SS];
    __shared__ float s_mxu[2][SS], s_mxo[2][SS], s_myu[2][SS], s_myo[2][SS];
    __shared__ float s_tw[SS * SS * 4];              // per-tap weight (incl. 1/4 pooling)
    __shared__ float s_T[2][NTAP_CHUNK][TROW];       // double-buffered taps x channels

    int box = blockIdx.x;
    if (box >= K) return;
    int tid = threadIdx.x;

    // ---- per-box scalars (redundant per thread; scalar loads) ----
    const float* ph = bxh + (size_t)box * 5;
    const float* po = bxo + (size_t)box * 5;
    int   bidx = (int)ph[0];
    float hx1 = ph[1], hy1 = ph[2], hx2 = ph[3], hy2 = ph[4];
    float ox1 = po[1], oy1 = po[2], ox2 = po[3], oy2 = po[4];
    float ux1 = fminf(hx1, ox1), uy1 = fminf(hy1, oy1);
    float ux2 = fmaxf(hx2, ox2), uy2 = fmaxf(hy2, oy2);
    float su = sqrtf((ux2 - ux1) * (uy2 - uy1));
    float so = sqrtf((ox2 - ox1) * (oy2 - oy1));
    float sm = fminf(su, so);
    float lvf = floorf(4.0f + log2f(sm / 224.0f + 1e-6f));
    lvf = fminf(fmaxf(lvf, 2.0f), 5.0f);
    int lvl = (int)lvf - 2;                               // 0..3
    const float* feat = (lvl == 0) ? f0 : (lvl == 1) ? f1 : (lvl == 2) ? f2 : f3;
    int   W     = 128 >> lvl;
    float scale = 0.25f / (float)(1 << lvl);
    float fs    = (float)W;

    float ubx1 = ux1 * scale, uby1 = uy1 * scale, ubx2 = ux2 * scale, uby2 = uy2 * scale;
    float obx1 = ox1 * scale, oby1 = oy1 * scale, obx2 = ox2 * scale, oby2 = oy2 * scale;
    float cux1 = fminf(fmaxf(ubx1, 0.f), fs), cuy1 = fminf(fmaxf(uby1, 0.f), fs);
    float cux2 = fminf(fmaxf(ubx2, 0.f), fs), cuy2 = fminf(fmaxf(uby2, 0.f), fs);
    float cox1 = fminf(fmaxf(obx1, 0.f), fs), coy1 = fminf(fmaxf(oby1, 0.f), fs);
    float cox2 = fminf(fmaxf(obx2, 0.f), fs), coy2 = fminf(fmaxf(oby2, 0.f), fs);

    // ---- sampling geometry (threads 0..13) ----
    if (tid < SS) {
        int i = tid;
        float t = ((float)i + 0.5f) * 0.5f;               // (i//2) + ((i%2)+0.5)/2
        float roi_w = fmaxf(ubx2 - ubx1, 1.0f);
        float roi_h = fmaxf(uby2 - uby1, 1.0f);
        float px = ubx1 + t * roi_w * (1.0f / 7.0f);
        float py = uby1 + t * roi_h * (1.0f / 7.0f);
        int xl, xh, yl, yh; float wxl, wxh, wyl, wyh;
        axis_interp(px, fs, xl, xh, wxl, wxh);
        axis_interp(py, fs, yl, yh, wyl, wyh);
        s_xlo[i] = xl; s_xhi[i] = xh; s_ylo[i] = yl; s_yhi[i] = yh;
        s_wx[0][i] = wxl; s_wx[1][i] = wxh;
        s_wy[0][i] = wyl; s_wy[1][i] = wyh;
        s_mxu[0][i] = coverage_at(cux1, cux2, (float)xl);
        s_mxu[1][i] = coverage_at(cux1, cux2, (float)xh);
        s_mxo[0][i] = coverage_at(cox1, cox2, (float)xl);
        s_mxo[1][i] = coverage_at(cox1, cox2, (float)xh);
        s_myu[0][i] = coverage_at(cuy1, cuy2, (float)yl);
        s_myu[1][i] = coverage_at(cuy1, cuy2, (float)yh);
        s_myo[0][i] = coverage_at(coy1, coy2, (float)yl);
        s_myo[1][i] = coverage_at(coy1, coy2, (float)yh);
    }
    __syncthreads();

    // ---- per-tap combined weight table: tw[(sy*14+sx)*4 + a*2+b] ----
    for (int sidx = tid; sidx < SS * SS * 4; sidx += 256) {
        int ab = sidx & 3;
        int b2 = ab & 1;
        int a2 = ab >> 1;
        int smp = sidx >> 2;
        int ssx = smp % SS;
        int ssy = smp / SS;
        float m = fmaxf(s_myu[a2][ssy] * s_mxu[b2][ssx],
                        s_myo[a2][ssy] * s_mxo[b2][ssx]);
        s_tw[sidx] = 0.25f * s_wy[a2][ssy] * s_wx[b2][ssx] * m;  // 0.25 = SRxSR mean
    }
    __syncthreads();

    // ---- GEMM D[64x256] = W[64x784] x T[784x256] via V_WMMA_F32_16X16X4_F32 ----
    int w  = tid >> 5;            // wave 0..7
    int l  = tid & 31;            // lane
    int mt = w >> 1;              // M-tile (16 output pixels)
    int nhalf = w & 1;            // channel half (128 channels)
    int lane16 = l & 15;
    int khalf  = (l < 16) ? 0 : 2;

    v8f acc[8] = {};              // 8 N-tiles of 16 channels

    const size_t HW = (size_t)W * (size_t)W;
    const float* fc = feat + ((size_t)bidx * CCH + (size_t)tid) * HW;  // this thread's channel

    // async gather of one 28-tap chunk into LDS buffer b (one instr per tap, ASYNCcnt +28)
    auto issue_chunk = [&](int c, int b) {
        int sy  = c >> 1;
        int sxb = (c & 1) * 7;
        int y0 = s_ylo[sy], y1i = s_yhi[sy];
        #pragma unroll 4
        for (int t = 0; t < NTAP_CHUNK; ++t) {
            int sx = sxb + (t >> 2);
            int yy = ((t >> 1) & 1) ? y1i : y0;
            int xx = (t & 1) ? s_xhi[sx] : s_xlo[sx];
            unsigned long long ga =
                (unsigned long long)(uintptr_t)(fc + (size_t)yy * W + xx);
            unsigned int la = (unsigned int)(uintptr_t)&s_T[b][t][tid];  // LDS byte offset
            asm volatile("global_load_async_to_lds_b32 %0, %1, off"
                         :: "v"(la), "v"(ga) : "memory");
        }
    };

    issue_chunk(0, 0);                       // prologue: chunk 0 -> buf 0

    for (int ch = 0; ch < 28; ++ch) {
        int buf = ch & 1;
        __syncthreads();                     // all waves done reading buf^1 (chunk ch-1)
        if (ch + 1 < 28) {
            issue_chunk(ch + 1, buf ^ 1);    // overlap next gather with this compute
            asm volatile("s_wait_asynccnt 28" ::: "memory");  // chunk ch complete (in-order)
        } else {
            asm volatile("s_wait_asynccnt 0" ::: "memory");
        }
        __syncthreads();                     // chunk ch visible block-wide

        int sy  = ch >> 1;
        int sxb = (ch & 1) * 7;
        #pragma unroll
        for (int kk = 0; kk < 7; ++kk) {
            int sx = sxb + kk;
            int rstar = (sy >> 1) * 7 + (sx >> 1);   // only live output pixel this K-step
            if ((rstar >> 4) != mt) continue;        // wave-uniform skip (EXEC stays full)
            // A fragment (16x4 fp32): lanes 0-15 hold K=0/1, lanes 16-31 K=2/3
            float a0 = 0.0f, a1 = 0.0f;
            if (mt * 16 + lane16 == rstar) {
                int base = (sy * SS + sx) * 4 + khalf;
                a0 = s_tw[base];
                a1 = s_tw[base + 1];
            }
            v2f A; A.x = a0; A.y = a1;
            int trow = kk * 4 + khalf;
            #pragma unroll
            for (int j = 0; j < 8; ++j) {
                int cb = (nhalf * 8 + j) * 16 + lane16;
                v2f B; B.x = s_T[buf][trow][cb]; B.y = s_T[buf][trow + 1][cb];
                acc[j] = __builtin_amdgcn_wmma_f32_16x16x4_f32(
                    false, A, false, B, (short)0, acc[j], false, false);
            }
        }
    }

    // ---- store D tiles: out[box][c][p], p = M row, c = N col ----
    size_t obase = (size_t)box * CCH * 49;
    int mhi = (l >= 16) ? 8 : 0;
    #pragma unroll
    for (int j = 0; j < 8; ++j) {
        int cc = (nhalf * 8 + j) * 16 + lane16;
        #pragma unroll
        for (int v = 0; v < 8; ++v) {
            int r = mt * 16 + v + mhi;
            if (r < 49) out[obase + (size_t)cc * 49 + r] = acc[j][v];
        }
    }
}

extern "C" void kernel_launch(void* const* d_in, const int* in_sizes, int n_in,
                              void* d_out, int out_size, void* d_ws, size_t ws_size,
                              hipStream_t stream) {
    const float* f0 = (const float*)d_in[0];
    const float* f1 = (const float*)d_in[1];
    const float* f2 = (const float*)d_in[2];
    const float* f3 = (const float*)d_in[3];
    const float* bh = (const float*)d_in[4];
    const float* bo = (const float*)d_in[5];
    float* out = (float*)d_out;
    int K = in_sizes[4] / 5;   // boxes_h is [K,5]
    dim3 grid(K), block(256);
    boxpair_roialign_wmma<<<grid, block, 0, stream>>>(f0, f1, f2, f3, bh, bo, out, K);
}